// HeteroCritic_67551245631994
// MI455X (gfx1250) — compile-verified
//
#include <hip/hip_runtime.h>
#include <math.h>

#define DIN 128
#define HID 16

typedef __attribute__((ext_vector_type(2))) float v2f;
typedef __attribute__((ext_vector_type(8))) float v8f;

// ---------- ordered-uint mapping for float atomic max ----------
__device__ __forceinline__ unsigned fkey(float f) {
    unsigned b = __float_as_uint(f);
    return (b & 0x80000000u) ? ~b : (b | 0x80000000u);
}
__device__ __forceinline__ float funkey(unsigned k) {
    unsigned b = (k & 0x80000000u) ? (k & 0x7FFFFFFFu) : ~k;
    return __uint_as_float(b);
}
__device__ __forceinline__ float lrelu02(float a) { return a > 0.f ? a : 0.2f * a; }

// ---------- Kernel 1: fp32 projection via V_WMMA_F32_16X16X4_F32 ----------
// h[n,16] = x[n,128] @ w[128,16] + b[16].  One wave = one 16x16 output tile.
// W is staged in LDS in B-fragment layout; A rows are clamped (branch-free loads,
// out-of-range rows only affect discarded D rows).
__global__ void k_proj_wmma(const float* __restrict__ x, const float* __restrict__ w,
                            const float* __restrict__ bias, float* __restrict__ h, int n) {
    __shared__ float wlds[DIN / 4][32][2];     // [kc][lane][2] = 8 KB
    const int lane = threadIdx.x & 31;
    const int wave = threadIdx.x >> 5;
    const int half = lane >> 4;                // 0: K pair {0,1}; 1: K pair {2,3}
    const int l16  = lane & 15;

    // cooperative fill of B fragments: wlds[kc][ln] = { w[k0][col], w[k0+1][col] }
    for (int idx = threadIdx.x; idx < (DIN / 4) * 32; idx += blockDim.x) {
        const int kc = idx >> 5, ln = idx & 31;
        const int k0 = kc * 4 + (ln >> 4) * 2, col = ln & 15;
        wlds[kc][ln][0] = w[(k0)     * HID + col];
        wlds[kc][ln][1] = w[(k0 + 1) * HID + col];
    }
    __syncthreads();

    const int tile = blockIdx.x * (blockDim.x >> 5) + wave;
    if (tile >= (n + 15) >> 4) return;

    int row = tile * 16 + l16;
    if (row >= n) row = n - 1;                 // clamp: branch-free A loads
    const v2f* __restrict__ xa = (const v2f*)(x + (size_t)row * DIN + half * 2);

    v8f c = {};
    #pragma unroll
    for (int kc = 0; kc < DIN / 4; ++kc) {
        const v2f a = xa[kc * 2];                              // global_load_b64
        const v2f b = *(const v2f*)&wlds[kc][lane][0];         // ds_load_b64
        c = __builtin_amdgcn_wmma_f32_16x16x4_f32(false, a, false, b, (short)0, c, false, false);
    }

    const float bn = bias[l16];
    if (tile * 16 + 16 <= n) {                 // full tile: unguarded stores
        #pragma unroll
        for (int j = 0; j < 8; ++j) {
            const int r = tile * 16 + j + (half ? 8 : 0);
            h[(size_t)r * HID + l16] = c[j] + bn;
        }
    } else {
        #pragma unroll
        for (int j = 0; j < 8; ++j) {
            const int r = tile * 16 + j + (half ? 8 : 0);
            if (r < n) h[(size_t)r * HID + l16] = c[j] + bn;
        }
    }
}

// ---------- Kernel 2: per-node attention scalars (8 dot products) ----------
__global__ void k_alpha(const float* __restrict__ hp, const float* __restrict__ ht,
                        const float* __restrict__ a_src_pt, const float* __restrict__ a_dst_pt,
                        const float* __restrict__ a_src_tp, const float* __restrict__ a_dst_tp,
                        const float* __restrict__ a_src_pp, const float* __restrict__ a_dst_pp,
                        const float* __restrict__ a_src_tt, const float* __restrict__ a_dst_tt,
                        float* __restrict__ s_pt, float* __restrict__ d_pt,
                        float* __restrict__ s_tp, float* __restrict__ d_tp,
                        float* __restrict__ s_pp, float* __restrict__ d_pp,
                        float* __restrict__ s_tt, float* __restrict__ d_tt, int n) {
    const int i = blockIdx.x * blockDim.x + threadIdx.x;
    if (i >= n) return;
    const float4* __restrict__ hp4 = (const float4*)(hp + (size_t)i * HID);
    const float4* __restrict__ ht4 = (const float4*)(ht + (size_t)i * HID);
    float vs_pt = 0.f, vd_tp = 0.f, vs_pp = 0.f, vd_pp = 0.f;
    float vd_pt = 0.f, vs_tp = 0.f, vs_tt = 0.f, vd_tt = 0.f;
    #pragma unroll
    for (int j = 0; j < HID / 4; ++j) {
        const float4 p = hp4[j];
        const float4 t = ht4[j];
        const float pe[4] = { p.x, p.y, p.z, p.w };
        const float te[4] = { t.x, t.y, t.z, t.w };
        #pragma unroll
        for (int k = 0; k < 4; ++k) {
            const int hcol = j * 4 + k;
            vs_pt += pe[k] * a_src_pt[hcol];  vd_tp += pe[k] * a_dst_tp[hcol];
            vs_pp += pe[k] * a_src_pp[hcol];  vd_pp += pe[k] * a_dst_pp[hcol];
            vd_pt += te[k] * a_dst_pt[hcol];  vs_tp += te[k] * a_src_tp[hcol];
            vs_tt += te[k] * a_src_tt[hcol];  vd_tt += te[k] * a_dst_tt[hcol];
        }
    }
    s_pt[i] = vs_pt; d_pt[i] = vd_pt; s_tp[i] = vs_tp; d_tp[i] = vd_tp;
    s_pp[i] = vs_pp; d_pp[i] = vd_pp; s_tt[i] = vs_tt; d_tt[i] = vd_tt;
}

// ---------- Kernel 3a: segment max of leaky-relu logits ----------
__global__ void k_edge_max(const int* __restrict__ src, const int* __restrict__ dst,
                           const float* __restrict__ s, const float* __restrict__ d,
                           unsigned* __restrict__ mkeys, int E) {
    const int e = blockIdx.x * blockDim.x + threadIdx.x;
    if (e >= E) return;
    if (e + 4096 < E) {           // lowers to global_prefetch_b8 on gfx1250
        __builtin_prefetch(src + e + 4096, 0, 0);
        __builtin_prefetch(dst + e + 4096, 0, 0);
    }
    const int si = src[e], di = dst[e];
    const float a = lrelu02(s[si] + d[di]);
    atomicMax(&mkeys[di], fkey(a));
}

// ---------- Kernel 3b: segment sum of exp(logit - max) ----------
__global__ void k_edge_sum(const int* __restrict__ src, const int* __restrict__ dst,
                           const float* __restrict__ s, const float* __restrict__ d,
                           const unsigned* __restrict__ mkeys, float* __restrict__ sum, int E) {
    const int e = blockIdx.x * blockDim.x + threadIdx.x;
    if (e >= E) return;
    const int si = src[e], di = dst[e];
    const float a = lrelu02(s[si] + d[di]);
    atomicAdd(&sum[di], __expf(a - funkey(mkeys[di])));
}

// ---------- Kernel 3c: weighted scatter-add of h_src rows (16 lanes/edge) ----------
__global__ void k_edge_scatter(const int* __restrict__ src, const int* __restrict__ dst,
                               const float* __restrict__ s, const float* __restrict__ d,
                               const unsigned* __restrict__ mkeys, const float* __restrict__ sum,
                               const float* __restrict__ hsrc, float* __restrict__ out,
                               long long E16) {
    const long long t = (long long)blockIdx.x * blockDim.x + threadIdx.x;
    if (t >= E16) return;
    const long long e = t >> 4;
    const int hcol = (int)(t & 15);
    const int si = src[e], di = dst[e];
    const float a = lrelu02(s[si] + d[di]);
    const float w = __expf(a - funkey(mkeys[di])) / (sum[di] + 1e-16f);
    atomicAdd(&out[(long long)di * HID + hcol], w * hsrc[(long long)si * HID + hcol]);
}

// ---------- Kernel 4: semantic reduce  t[m] += sum_i tanh(relu(out_m[i]) @ k_w + k_b) ----------
// One wave per 16-node tile; z @ k_w via 4 chained V_WMMA_F32_16X16X4_F32.
__global__ void k_sem_reduce(const float* __restrict__ outA, const float* __restrict__ outB,
                             const float* __restrict__ kw, const float* __restrict__ kb,
                             float* __restrict__ t /*[2*HID]*/, int n) {
    __shared__ float kwlds[HID / 4][32][2];    // 1 KB, B-fragment layout
    const int lane = threadIdx.x & 31;
    const int wave = threadIdx.x >> 5;
    const int half = lane >> 4;
    const int l16  = lane & 15;

    if (threadIdx.x < (HID / 4) * 32) {
        const int kc = threadIdx.x >> 5, ln = threadIdx.x & 31;
        const int k0 = kc * 4 + (ln >> 4) * 2, col = ln & 15;
        kwlds[kc][ln][0] = kw[(k0)     * HID + col];
        kwlds[kc][ln][1] = kw[(k0 + 1) * HID + col];
    }
    __syncthreads();

    const int tile = blockIdx.x * (blockDim.x >> 5) + wave;
    if (tile * 16 >= n) return;

    int row = tile * 16 + l16;
    if (row >= n) row = n - 1;                 // clamp: branch-free loads
    const float* outs[2] = { outA, outB };
    const float bn = kb[l16];

    for (int m = 0; m < 2; ++m) {
        const v2f* __restrict__ oa = (const v2f*)(outs[m] + (size_t)row * HID + half * 2);
        v8f c = {};
        #pragma unroll
        for (int kc = 0; kc < HID / 4; ++kc) {
            v2f a = oa[kc * 2];                                // global_load_b64
            a.x = fmaxf(a.x, 0.f);                             // relu -> z
            a.y = fmaxf(a.y, 0.f);
            const v2f b = *(const v2f*)&kwlds[kc][lane][0];    // ds_load_b64
            c = __builtin_amdgcn_wmma_f32_16x16x4_f32(false, a, false, b, (short)0, c, false, false);
        }
        float acc = 0.f;
        #pragma unroll
        for (int j = 0; j < 8; ++j) {
            const int r = tile * 16 + j + (half ? 8 : 0);
            if (r < n) acc += tanhf(c[j] + bn);
        }
        atomicAdd(&t[m * HID + l16], acc);
    }
}

// ---------- Kernel 5: semantic softmax scores (tiny) ----------
__global__ void k_score(const float* __restrict__ tP, const float* __restrict__ tT,
                        const float* __restrict__ q, float invn, float* __restrict__ attn) {
    if (threadIdx.x != 0 || blockIdx.x != 0) return;
    for (int ty = 0; ty < 2; ++ty) {
        const float* t = ty ? tT : tP;
        float s0 = 0.f, s1 = 0.f;
        for (int hcol = 0; hcol < HID; ++hcol) {
            s0 += q[hcol] * t[hcol] * invn;
            s1 += q[hcol] * t[HID + hcol] * invn;
        }
        const float mx = fmaxf(s0, s1);
        const float e0 = __expf(s0 - mx), e1 = __expf(s1 - mx);
        const float inv = 1.f / (e0 + e1);
        attn[ty * 2 + 0] = e0 * inv;
        attn[ty * 2 + 1] = e1 * inv;
    }
}

// ---------- Kernel 6: pooled[h] += sum_i relu(a0*relu(outA)+a1*relu(outB)) ----------
__global__ void k_pool(const float* __restrict__ outA, const float* __restrict__ outB,
                       const float* __restrict__ attn, float* __restrict__ pooled, int n) {
    __shared__ float sbuf[256];
    const int tid = threadIdx.x;
    const int hcol = tid & 15;
    const float a0 = attn[0], a1 = attn[1];
    float acc = 0.f;
    for (long long r = (long long)blockIdx.x * 16 + (tid >> 4); r < n;
         r += (long long)gridDim.x * 16) {
        const float z0 = fmaxf(outA[r * HID + hcol], 0.f);
        const float z1 = fmaxf(outB[r * HID + hcol], 0.f);
        acc += fmaxf(a0 * z0 + a1 * z1, 0.f);
    }
    sbuf[tid] = acc;
    __syncthreads();
    if (tid < 16) {
        float s = 0.f;
        #pragma unroll
        for (int k = 0; k < 16; ++k) s += sbuf[tid + 16 * k];
        atomicAdd(&pooled[tid], s);
    }
}

// ---------- Kernel 7: final linear ----------
__global__ void k_final(const float* __restrict__ pooled, const float* __restrict__ lin_w,
                        const float* __restrict__ lin_b, float* __restrict__ out) {
    if (threadIdx.x != 0 || blockIdx.x != 0) return;
    float acc = lin_b[0];
    #pragma unroll
    for (int hcol = 0; hcol < HID; ++hcol) acc += pooled[hcol] * lin_w[hcol];
    out[0] = acc;
}

extern "C" void kernel_launch(void* const* d_in, const int* in_sizes, int n_in,
                              void* d_out, int out_size, void* d_ws, size_t ws_size,
                              hipStream_t stream) {
    const float* x_place = (const float*)d_in[0];
    const float* x_trans = (const float*)d_in[1];
    const float* w_p = (const float*)d_in[2];  const float* b_p = (const float*)d_in[3];
    const float* w_t = (const float*)d_in[4];  const float* b_t = (const float*)d_in[5];
    const float* a_src_pt = (const float*)d_in[6];  const float* a_dst_pt = (const float*)d_in[7];
    const float* a_src_tp = (const float*)d_in[8];  const float* a_dst_tp = (const float*)d_in[9];
    const float* a_src_pp = (const float*)d_in[10]; const float* a_dst_pp = (const float*)d_in[11];
    const float* a_src_tt = (const float*)d_in[12]; const float* a_dst_tt = (const float*)d_in[13];
    const float* q    = (const float*)d_in[14];
    const float* k_w  = (const float*)d_in[15];
    const float* k_b  = (const float*)d_in[16];
    const float* linw = (const float*)d_in[17];
    const float* linb = (const float*)d_in[18];
    const int* edge_pt = (const int*)d_in[19];
    const int* edge_tp = (const int*)d_in[20];
    const int* edge_pp = (const int*)d_in[21];
    const int* edge_tt = (const int*)d_in[22];

    const int n  = in_sizes[0] / DIN;       // nodes per type (100000)
    const int E  = in_sizes[19] / 2;        // edges per type (3.2M)

    // ---- workspace layout (floats) ----
    float* ws = (float*)d_ws;
    size_t o = 0;
    float* hp = ws + o; o += (size_t)n * HID;
    float* ht = ws + o; o += (size_t)n * HID;
    float* alpha[8];                         // s_pt,d_pt,s_tp,d_tp,s_pp,d_pp,s_tt,d_tt
    for (int i = 0; i < 8; ++i) { alpha[i] = ws + o; o += (size_t)n; }
    const size_t zstart = o;                 // zero-init region starts here
    unsigned* mkeys[4]; float* sums[4]; float* outs[4];
    for (int i = 0; i < 4; ++i) { mkeys[i] = (unsigned*)(ws + o); o += (size_t)n; }
    for (int i = 0; i < 4; ++i) { sums[i]  = ws + o;              o += (size_t)n; }
    for (int i = 0; i < 4; ++i) { outs[i]  = ws + o;              o += (size_t)n * HID; }
    float* tT     = ws + o; o += 2 * HID;    // trans-type semantic sums
    float* tP     = ws + o; o += 2 * HID;    // place-type semantic sums
    float* attn   = ws + o; o += 4;          // [p0,p1,t0,t1]
    float* pooled = ws + o; o += HID;
    hipMemsetAsync(ws + zstart, 0, (o - zstart) * sizeof(float), stream);

    // ---- 1. projections (WMMA) ----
    const int tiles = (n + 15) / 16;
    const int projBlocks = (tiles + 3) / 4;                 // 4 waves / 128-thread block
    k_proj_wmma<<<projBlocks, 128, 0, stream>>>(x_place, w_p, b_p, hp, n);
    k_proj_wmma<<<projBlocks, 128, 0, stream>>>(x_trans, w_t, b_t, ht, n);

    // ---- 2. per-node attention scalars ----
    k_alpha<<<(n + 255) / 256, 256, 0, stream>>>(hp, ht,
        a_src_pt, a_dst_pt, a_src_tp, a_dst_tp, a_src_pp, a_dst_pp, a_src_tt, a_dst_tt,
        alpha[0], alpha[1], alpha[2], alpha[3], alpha[4], alpha[5], alpha[6], alpha[7], n);

    // ---- 3. edge attention: {edges, s, d, hsrc, slot} per edge type ----
    struct ET { const int* e; const float* s; const float* d; const float* hs; int slot; };
    const ET ets[4] = {
        { edge_pt, alpha[0], alpha[1], hp, 0 },   // place->trans  (out_t[0])
        { edge_tt, alpha[6], alpha[7], ht, 1 },   // trans->trans  (out_t[1])
        { edge_tp, alpha[2], alpha[3], ht, 2 },   // trans->place  (out_p[0])
        { edge_pp, alpha[4], alpha[5], hp, 3 },   // place->place  (out_p[1])
    };
    const int eBlocks = (E + 255) / 256;
    const long long E16 = (long long)E * HID;
    const int sBlocks = (int)((E16 + 255) / 256);
    for (int i = 0; i < 4; ++i) {
        const ET& et = ets[i];
        const int* src = et.e;          // edge[0,:]
        const int* dst = et.e + E;      // edge[1,:]
        k_edge_max<<<eBlocks, 256, 0, stream>>>(src, dst, et.s, et.d, mkeys[et.slot], E);
        k_edge_sum<<<eBlocks, 256, 0, stream>>>(src, dst, et.s, et.d, mkeys[et.slot],
                                                sums[et.slot], E);
        k_edge_scatter<<<sBlocks, 256, 0, stream>>>(src, dst, et.s, et.d, mkeys[et.slot],
                                                    sums[et.slot], et.hs, outs[et.slot], E16);
    }

    // ---- 4. semantic attention (WMMA) ----
    k_sem_reduce<<<projBlocks, 128, 0, stream>>>(outs[0], outs[1], k_w, k_b, tT, n); // trans
    k_sem_reduce<<<projBlocks, 128, 0, stream>>>(outs[2], outs[3], k_w, k_b, tP, n); // place
    k_score<<<1, 32, 0, stream>>>(tP, tT, q, 1.0f / (float)n, attn);

    // ---- 5. pooled sums and final linear ----
    k_pool<<<512, 256, 0, stream>>>(outs[2], outs[3], attn + 0, pooled, n);  // place
    k_pool<<<512, 256, 0, stream>>>(outs[0], outs[1], attn + 2, pooled, n);  // trans
    k_final<<<1, 32, 0, stream>>>(pooled, linw, linb, (float*)d_out);
}